// VGCL_48687749268104
// MI455X (gfx1250) — compile-verified
//
#include <hip/hip_runtime.h>

#define NUM_USER 150000
#define NUM_ITEM 75000
#define NTOT (NUM_USER + NUM_ITEM)
#define D 64
#define NNZ 2000000
#define ND ((size_t)NTOT * D)          // 14,400,000 floats per [N,D] slab
#define ND4 ((NTOT * D) / 4)           // float4 count
#define SCALE 0.01f

typedef __attribute__((ext_vector_type(2))) float v2f;
typedef __attribute__((ext_vector_type(8))) float v8f;

// ---------------------------------------------------------------------------
// init: b0 = concat(user_emb, item_emb); b1 = 0; acc = 0   (float4 vectorized)
// ---------------------------------------------------------------------------
__global__ void vgcl_init(const float4* __restrict__ user, const float4* __restrict__ item,
                          float4* __restrict__ b0, float4* __restrict__ b1,
                          float4* __restrict__ acc) {
  int i = blockIdx.x * blockDim.x + threadIdx.x;
  if (i >= ND4) return;
  const int userN4 = NUM_USER * D / 4;
  float4 v = (i < userN4) ? user[i] : item[i - userN4];
  b0[i] = v;
  float4 z = make_float4(0.f, 0.f, 0.f, 0.f);
  b1[i] = z;
  acc[i] = z;
}

// ---------------------------------------------------------------------------
// SpMM: y[r,:] += vals[e] * x[c,:]  — one wave32 per edge, 2 features per lane
// (feature matrix is L2-resident at 57.6MB vs 192MB L2; scatter via f32 atomics)
// ---------------------------------------------------------------------------
__global__ void vgcl_spmm(const int* __restrict__ rows, const int* __restrict__ cols,
                          const float* __restrict__ vals, const float* __restrict__ x,
                          float* __restrict__ y) {
  int e = blockIdx.x * (blockDim.x >> 5) + (threadIdx.x >> 5);
  if (e >= NNZ) return;
  int lane = threadIdx.x & 31;
  int r = rows[e];
  int c = cols[e];
  float v = vals[e];
  float2 xv = *((const float2*)(x + (size_t)c * D) + lane);
  float* yp = y + (size_t)r * D + lane * 2;
  atomicAdd(yp + 0, v * xv.x);
  atomicAdd(yp + 1, v * xv.y);
}

// ---------------------------------------------------------------------------
// post-layer: acc += src; zero the buffer that becomes the next scatter target
// ---------------------------------------------------------------------------
__global__ void vgcl_post(float4* __restrict__ acc, const float4* __restrict__ src,
                          float4* __restrict__ zeroed) {
  int i = blockIdx.x * blockDim.x + threadIdx.x;
  if (i >= ND4) return;
  float4 a = acc[i];
  float4 s = src[i];
  a.x += s.x; a.y += s.y; a.z += s.z; a.w += s.w;
  acc[i] = a;
  zeroed[i] = make_float4(0.f, 0.f, 0.f, 0.f);
}

// ---------------------------------------------------------------------------
// finalize: mean = (acc + last_layer) / 3
// ---------------------------------------------------------------------------
__global__ void vgcl_finalize(const float4* __restrict__ acc, const float4* __restrict__ last,
                              float4* __restrict__ mean) {
  int i = blockIdx.x * blockDim.x + threadIdx.x;
  if (i >= ND4) return;
  float4 a = acc[i];
  float4 l = last[i];
  const float inv = 1.0f / 3.0f;
  float4 m = make_float4((a.x + l.x) * inv, (a.y + l.y) * inv,
                         (a.z + l.z) * inv, (a.w + l.w) * inv);
  mean[i] = m;
}

// ---------------------------------------------------------------------------
// WMMA kernel: logstd = mean @ W + bias via V_WMMA_F32_16X16X4_F32 (K=64 = 16
// chained WMMAs), then std = exp(logstd), view{1,2} = mean + std*noise*SCALE.
// One wave32 per 16-row tile; 4 N-tiles of 16 cols -> 64 WMMAs per wave.
// A frag (16x4 f32): lane m=lane&15 holds row m, k-pair {4j+2*half, +1}.
// B frag (4x16):     lane holds col n=lane&15, same k-pair.
// C/D:               VGPR i -> row M = i + 8*half, col N = lane&15.
// ---------------------------------------------------------------------------
__global__ void vgcl_gemm_views(const float* __restrict__ mean, const float* __restrict__ W,
                                const float* __restrict__ bias, const float* __restrict__ n1,
                                const float* __restrict__ n2, float* __restrict__ out) {
  const int numTiles = (NTOT + 15) / 16;
  int tile = blockIdx.x * (blockDim.x >> 5) + (threadIdx.x >> 5);
  if (tile >= numTiles) return;  // wave-uniform exit; surviving waves keep EXEC all-1s

  int lane = threadIdx.x & 31;
  int m = lane & 15;
  int half = lane >> 4;

  // Load A fragments (clamp tail rows; stores are masked below)
  int rowA = tile * 16 + m;
  if (rowA >= NTOT) rowA = NTOT - 1;
  const float* arow = mean + (size_t)rowA * D;
  v2f a[16];
#pragma unroll
  for (int j = 0; j < 16; ++j)
    a[j] = *(const v2f*)(arow + 4 * j + 2 * half);  // 8B-aligned (even offset)

#pragma unroll
  for (int nt = 0; nt < 4; ++nt) {
    int n = nt * 16 + m;
    float bn = bias[n];
    v8f c;
#pragma unroll
    for (int i = 0; i < 8; ++i) c[i] = bn;

#pragma unroll
    for (int j = 0; j < 16; ++j) {
      int k = 4 * j + 2 * half;
      v2f b;
      b.x = W[(size_t)k * D + n];
      b.y = W[(size_t)(k + 1) * D + n];
      // 8 args: (neg_a, A, neg_b, B, c_mod, C, reuse_a, reuse_b)
      c = __builtin_amdgcn_wmma_f32_16x16x4_f32(false, a[j], false, b,
                                                (short)0, c, false, false);
    }

#pragma unroll
    for (int i = 0; i < 8; ++i) {
      int r = tile * 16 + half * 8 + i;
      if (r < NTOT) {
        size_t off = (size_t)r * D + n;
        float sd = __expf(c[i]);
        float mv = mean[off];
        out[ND + off]     = sd;                             // std
        out[2 * ND + off] = mv + sd * n1[off] * SCALE;      // view1
        out[3 * ND + off] = mv + sd * n2[off] * SCALE;      // view2
      }
    }
  }
}

// ---------------------------------------------------------------------------
// Launch: scratch lives in the not-yet-final output slots (no ws needed).
//   slot0 = mean (final), slot1 = acc scratch -> std, slot2 = b1 -> view1,
//   slot3 = b0 -> view2.
// ---------------------------------------------------------------------------
extern "C" void kernel_launch(void* const* d_in, const int* in_sizes, int n_in,
                              void* d_out, int out_size, void* d_ws, size_t ws_size,
                              hipStream_t stream) {
  const float* user = (const float*)d_in[0];
  const float* item = (const float*)d_in[1];
  const float* W    = (const float*)d_in[2];
  const float* bias = (const float*)d_in[3];
  const int*   rows = (const int*)d_in[4];
  const int*   cols = (const int*)d_in[5];
  const float* vals = (const float*)d_in[6];
  const float* n1   = (const float*)d_in[7];
  const float* n2   = (const float*)d_in[8];
  float* out = (float*)d_out;

  float* meanBuf = out;            // final mean
  float* accBuf  = out + ND;       // acc scratch, later overwritten with std
  float* b1      = out + 2 * ND;   // ping, later view1
  float* b0      = out + 3 * ND;   // pong, later view2

  const int ewBlocks   = (ND4 + 255) / 256;
  const int spmmBlocks = (NNZ + 7) / 8;          // 8 waves (edges) per 256-thread block

  vgcl_init<<<ewBlocks, 256, 0, stream>>>((const float4*)user, (const float4*)item,
                                          (float4*)b0, (float4*)b1, (float4*)accBuf);
  // layer 1: b0 -> b1
  vgcl_spmm<<<spmmBlocks, 256, 0, stream>>>(rows, cols, vals, b0, b1);
  vgcl_post<<<ewBlocks, 256, 0, stream>>>((float4*)accBuf, (const float4*)b1, (float4*)b0);
  // layer 2: b1 -> b0
  vgcl_spmm<<<spmmBlocks, 256, 0, stream>>>(rows, cols, vals, b1, b0);
  vgcl_post<<<ewBlocks, 256, 0, stream>>>((float4*)accBuf, (const float4*)b0, (float4*)b1);
  // layer 3: b0 -> b1
  vgcl_spmm<<<spmmBlocks, 256, 0, stream>>>(rows, cols, vals, b0, b1);
  // mean = (acc + layer3) / 3
  vgcl_finalize<<<ewBlocks, 256, 0, stream>>>((const float4*)accBuf, (const float4*)b1,
                                              (float4*)meanBuf);
  // dense GEMM + exp + views (WMMA)
  const int numTiles = (NTOT + 15) / 16;
  const int gvBlocks = (numTiles + 7) / 8;       // 8 waves (tiles) per block
  vgcl_gemm_views<<<gvBlocks, 256, 0, stream>>>(meanBuf, W, bias, n1, n2, out);
}